// AltDiffLayer_52493090292022
// MI455X (gfx1250) — compile-verified
//
#include <hip/hip_runtime.h>

// Problem dims (from reference)
#define BATCH 64
#define NN    128
#define MEQ   32
#define DIN   64
#define RHO_F 1.0f
#define THRES_F 1e-5f
#define MAX_IT 5000
#define PITCH 129   // 128 + 1 padding: (i*129+k) % 64 == (i+k) % 64 -> conflict-free both ways

typedef __attribute__((ext_vector_type(2))) float v2f;
typedef __attribute__((ext_vector_type(8))) float v8f;

// ---------------------------------------------------------------------------
// Kernel A: M = Q + RHO*(A^T A + G^T G) per batch, built with f32 WMMA.
// grid = (BATCH, 8 tile-rows), block = 256 threads = 8 waves; wave w owns the
// 16x16 output tile (row = blockIdx.y, col = w). K stepped by 4 per
// V_WMMA_F32_16X16X4_F32.
//
// f32 16x16x4 lane layout (ISA 7.12.2):
//   A (16x4): lane L holds A[L&15][2*(L>>4)] in v0, A[L&15][2*(L>>4)+1] in v1
//   B (4x16): lane L holds B[2*(L>>4)][L&15] in v0, B[2*(L>>4)+1][L&15] in v1
//   C/D     : vgpr r, lane L -> row = r + 8*(L>>4), col = L&15
// ---------------------------------------------------------------------------
__global__ __launch_bounds__(256) void build_M_wmma(
    const float* __restrict__ Q, const float* __restrict__ A,
    const float* __restrict__ G, float* __restrict__ Mout) {
  const int b    = blockIdx.x;
  const int ti   = blockIdx.y;           // tile row 0..7
  const int wave = threadIdx.x >> 5;     // tile col 0..7
  const int lane = threadIdx.x & 31;
  const int r    = lane & 15;
  const int ks   = (lane >> 4) * 2;      // K sub-offset (0 or 2)
  const int arow = ti * 16 + r;          // output row index fed to A operand
  const int bcol = wave * 16 + r;        // output col index fed to B operand

  const float* Ab = A + b * (MEQ * NN);
  const float* Gb = G + b * (DIN * NN);

  v8f acc = {0.f, 0.f, 0.f, 0.f, 0.f, 0.f, 0.f, 0.f};

  // A^T A : out[i][j] = sum_m A[m][i]*A[m][j], K = MEQ = 32
#pragma unroll
  for (int k0 = 0; k0 < MEQ; k0 += 4) {
    v2f av, bv;
    av.x = Ab[(k0 + ks) * NN + arow];
    av.y = Ab[(k0 + ks + 1) * NN + arow];
    bv.x = Ab[(k0 + ks) * NN + bcol];
    bv.y = Ab[(k0 + ks + 1) * NN + bcol];
    acc = __builtin_amdgcn_wmma_f32_16x16x4_f32(false, av, false, bv,
                                                (short)0, acc, false, false);
  }
  // G^T G : K = DIN = 64
#pragma unroll
  for (int k0 = 0; k0 < DIN; k0 += 4) {
    v2f av, bv;
    av.x = Gb[(k0 + ks) * NN + arow];
    av.y = Gb[(k0 + ks + 1) * NN + arow];
    bv.x = Gb[(k0 + ks) * NN + bcol];
    bv.y = Gb[(k0 + ks + 1) * NN + bcol];
    acc = __builtin_amdgcn_wmma_f32_16x16x4_f32(false, av, false, bv,
                                                (short)0, acc, false, false);
  }

  // M = Q + RHO*acc  (C/D layout: vgpr rr, lane -> row rr+8*(lane>>4), col lane&15)
  const float* Qb = Q + b * (NN * NN);
  float* Mb = Mout + b * (NN * NN);
#pragma unroll
  for (int rr = 0; rr < 8; ++rr) {
    const int orow = ti * 16 + rr + 8 * (lane >> 4);
    const int ocol = wave * 16 + (lane & 15);
    Mb[orow * NN + ocol] = Qb[orow * NN + ocol] + RHO_F * acc[rr];
  }
}

// ---------------------------------------------------------------------------
// Kernel B: in-place Gauss-Jordan inverse in LDS (SPD, no pivoting),
// writes R = -inv(M) back to ws.  One block (256 thr) per batch.
// ---------------------------------------------------------------------------
__global__ __launch_bounds__(256) void invert_M(float* __restrict__ Mws) {
  extern __shared__ float sM[];          // NN * PITCH floats
  __shared__ float s_ip;
  const int b = blockIdx.x;
  const int tid = threadIdx.x;
  float* Mb = Mws + b * NN * NN;

  for (int idx = tid; idx < NN * NN; idx += 256)
    sM[(idx >> 7) * PITCH + (idx & 127)] = Mb[idx];
  __syncthreads();

  const int i  = tid >> 1;               // row owned by this thread pair
  const int j0 = (tid & 1) * 64;         // column half

  for (int k = 0; k < NN; ++k) {
    if (tid == 0) s_ip = 1.0f / sM[k * PITCH + k];
    __syncthreads();
    const float ip = s_ip;
    if (tid < NN)                         // scale pivot row; A[k][k] -> 1/p
      sM[k * PITCH + tid] = (tid == k) ? ip : sM[k * PITCH + tid] * ip;
    __syncthreads();
    const float f = sM[i * PITCH + k];    // multiplier (read before writes)
    __syncthreads();
    if (i != k) {
      for (int jj = j0; jj < j0 + 64; ++jj) {
        const float base = (jj == k) ? 0.0f : sM[i * PITCH + jj];
        sM[i * PITCH + jj] = base - f * sM[k * PITCH + jj];
      }
    }
    __syncthreads();
  }

  for (int idx = tid; idx < NN * NN; idx += 256)
    Mb[idx] = -sM[(idx >> 7) * PITCH + (idx & 127)];   // R = -inv(M)
}

// ---------------------------------------------------------------------------
// Kernel C: persistent ADMM loop, one block per batch, everything in LDS.
// ---------------------------------------------------------------------------
__global__ __launch_bounds__(256) void admm_solve(
    const float* __restrict__ Rws, const float* __restrict__ Q,
    const float* __restrict__ qv, const float* __restrict__ G,
    const float* __restrict__ h,  const float* __restrict__ A,
    const float* __restrict__ bvec, float* __restrict__ out) {
  extern __shared__ float sm[];
  __shared__ float s_res;                 // res_cur (init -100, per reference)
  __shared__ int   s_done;

  float* lR  = sm;                        // 128*129
  float* lQ  = lR + NN * PITCH;           // 128*129
  float* lG  = lQ + NN * PITCH;           // 64*129
  float* lA  = lG + DIN * PITCH;          // 32*129
  float* x   = lA + MEQ * PITCH;          // 128
  float* v   = x + NN;                    // 128
  float* c0  = v + NN;                    // 128
  float* sk  = c0 + NN;                   // 64
  float* nu  = sk + DIN;                  // 64
  float* gx  = nu + DIN;                  // 64
  float* lam = gx + DIN;                  // 32
  float* ax  = lam + MEQ;                 // 32
  float* lh  = ax + MEQ;                  // 64
  float* lb  = lh + DIN;                  // 32
  float* lq  = lb + MEQ;                  // 128
  float* red = lq + NN;                   // 256

  const int b   = blockIdx.x;
  const int tid = threadIdx.x;

  // ---- stage everything into LDS (pitch 129 kills bank conflicts) ----
  for (int idx = tid; idx < NN * NN; idx += 256) {
    lR[(idx >> 7) * PITCH + (idx & 127)] = Rws[b * NN * NN + idx];
    lQ[(idx >> 7) * PITCH + (idx & 127)] = Q[b * NN * NN + idx];
  }
  for (int idx = tid; idx < DIN * NN; idx += 256)
    lG[(idx >> 7) * PITCH + (idx & 127)] = G[b * DIN * NN + idx];
  for (int idx = tid; idx < MEQ * NN; idx += 256)
    lA[(idx >> 7) * PITCH + (idx & 127)] = A[b * MEQ * NN + idx];
  if (tid < NN)  lq[tid] = qv[b * NN + tid];
  if (tid < DIN) { lh[tid] = h[b * DIN + tid]; sk[tid] = 0.f; nu[tid] = 0.f; }
  if (tid < MEQ) { lb[tid] = bvec[b * MEQ + tid]; lam[tid] = 0.f; }
  if (tid == 0)  { s_res = -100.0f; s_done = 0; }
  __syncthreads();

  // c0 = q - RHO*A^T b - RHO*G^T h   (constant part of the x-update RHS)
  if (tid < NN) {
    float acc = lq[tid];
    for (int m = 0; m < MEQ; ++m) acc -= RHO_F * lA[m * PITCH + tid] * lb[m];
    for (int d = 0; d < DIN; ++d) acc -= RHO_F * lG[d * PITCH + tid] * lh[d];
    c0[tid] = acc;
  }
  __syncthreads();

  const int row = tid & 127;
  const int kk0 = (tid >> 7) * 64;        // K half for split GEMV

  for (int it = 0; it < MAX_IT; ++it) {
    // v = c0 + A^T lam + G^T (nu + RHO*sk_prev)
    if (tid < NN) {
      float acc = c0[tid];
      for (int m = 0; m < MEQ; ++m) acc += lA[m * PITCH + tid] * lam[m];
      for (int d = 0; d < DIN; ++d) acc += lG[d * PITCH + tid] * (nu[d] + RHO_F * sk[d]);
      v[tid] = acc;
    }
    __syncthreads();

    // x = R v  (256 threads: 2 half-dots per row, combined via LDS)
    {
      float p = 0.f;
      for (int k = kk0; k < kk0 + 64; ++k) p += lR[row * PITCH + k] * v[k];
      red[tid] = p;
    }
    __syncthreads();
    if (tid < NN) x[tid] = red[tid] + red[tid + 128];
    __syncthreads();

    // Gx and Ax in parallel
    if (tid < DIN) {
      float p = 0.f;
      for (int n = 0; n < NN; ++n) p += lG[tid * PITCH + n] * x[n];
      gx[tid] = p;
    } else if (tid < DIN + MEQ) {
      const int m = tid - DIN;
      float p = 0.f;
      for (int n = 0; n < NN; ++n) p += lA[m * PITCH + n] * x[n];
      ax[m] = p;
    }
    __syncthreads();

    // sk = relu(h - Gx - nu/RHO); nu += RHO*(Gx + sk - h); lam += RHO*(Ax - b)
    if (tid < DIN) {
      float s = lh[tid] - gx[tid] - nu[tid] * (1.0f / RHO_F);
      s = s > 0.f ? s : 0.f;
      sk[tid] = s;
      nu[tid] += RHO_F * (gx[tid] + s - lh[tid]);
    } else if (tid < DIN + MEQ) {
      const int m = tid - DIN;
      lam[m] += RHO_F * (ax[m] - lb[m]);
    }
    __syncthreads();

    // res = sum_i x_i * (0.5*(Qx)_i + q_i)
    {
      float p = 0.f;
      for (int k = kk0; k < kk0 + 64; ++k) p += lQ[row * PITCH + k] * x[k];
      red[tid] = p;
    }
    __syncthreads();
    float contrib = 0.f;
    if (tid < NN) {
      const float y = red[tid] + red[tid + 128];
      contrib = x[tid] * (0.5f * y + lq[tid]);
    }
    __syncthreads();
    red[tid] = contrib;                   // zero for tid >= 128
    __syncthreads();
    for (int stride = 128; stride > 0; stride >>= 1) {
      if (tid < stride) red[tid] += red[tid + stride];
      __syncthreads();
    }
    if (tid == 0) {
      const float rp = s_res;             // previous res_cur
      const float rc = red[0];
      s_done = (fabsf((rc - rp) / rp) <= THRES_F) ? 1 : 0;
      s_res  = rc;
    }
    __syncthreads();
    if (s_done) break;                    // x of this iteration is retained
  }

  if (tid < NN) out[b * NN + tid] = x[tid];
}

// ---------------------------------------------------------------------------
extern "C" void kernel_launch(void* const* d_in, const int* in_sizes, int n_in,
                              void* d_out, int out_size, void* d_ws, size_t ws_size,
                              hipStream_t stream) {
  const float* Q  = (const float*)d_in[0];
  const float* q  = (const float*)d_in[1];
  const float* G  = (const float*)d_in[2];
  const float* h  = (const float*)d_in[3];
  const float* A  = (const float*)d_in[4];
  const float* b  = (const float*)d_in[5];
  float* out = (float*)d_out;
  float* Mws = (float*)d_ws;                       // BATCH*128*128 f32 = 4 MB

  build_M_wmma<<<dim3(BATCH, 8), 256, 0, stream>>>(Q, A, G, Mws);

  invert_M<<<dim3(BATCH), 256, (size_t)(NN * PITCH) * sizeof(float), stream>>>(Mws);

  const size_t smc =
      (size_t)(2 * NN * PITCH + DIN * PITCH + MEQ * PITCH   // R, Q, G, A
               + 3 * NN                                     // x, v, c0
               + 3 * DIN                                    // sk, nu, gx
               + 2 * MEQ                                    // lam, ax
               + DIN + MEQ + NN                             // h, b, q
               + 256) * sizeof(float);                      // reduction
  admm_solve<<<dim3(BATCH), 256, smc, stream>>>(Mws, Q, q, G, h, A, b, out);
}